// BertQueryNER_24833500905898
// MI455X (gfx1250) — compile-verified
//
#include <hip/hip_runtime.h>
#include <math.h>

typedef __attribute__((ext_vector_type(2))) float v2f;
typedef __attribute__((ext_vector_type(8))) float v8f;

#define Bdim 8
#define Ldim 128
#define Hdim 768

// ---------------------------------------------------------------------------
// Kernel 1: A' = X @ W1[0:H,:] + b1   (slab 0)
//           Bm = X @ W1[H:2H,:]       (slab 1)
// X: (1024, 768) row-major; W1: (1536, 768) row-major.
// One wave -> one 16x16 output tile via V_WMMA_F32_16X16X4_F32.
// Block = 256 threads = 8 waves arranged 2(M) x 4(N) -> 32x64 block tile.
// grid = (768/64, 1024/32, 2)
// ---------------------------------------------------------------------------
__global__ __launch_bounds__(256) void gemm_proj_wmma(
    const float* __restrict__ X, const float* __restrict__ W1,
    const float* __restrict__ b1, float* __restrict__ Aout,
    float* __restrict__ Bout)
{
  const int wave   = threadIdx.x >> 5;
  const int lane   = threadIdx.x & 31;
  const int laneLo = lane & 15;
  const int kHi    = (lane >> 4) << 1;          // 0 for lanes 0-15, 2 for 16-31
  const int tileM  = blockIdx.y * 32 + (wave & 1) * 16;
  const int tileN  = blockIdx.x * 64 + (wave >> 1) * 16;
  const int slab   = blockIdx.z;

  const float* __restrict__ W = W1 + (size_t)slab * Hdim * Hdim;
  const int m = tileM + laneLo;                 // A-frag row for this lane
  const int n = tileN + laneLo;                 // B-frag col for this lane
  const float* __restrict__ xrow = X + (size_t)m * Hdim;

  v8f c = {0.f, 0.f, 0.f, 0.f, 0.f, 0.f, 0.f, 0.f};

#pragma unroll 4
  for (int k0 = 0; k0 < Hdim; k0 += 4) {
    const int ka = k0 + kHi;                    // even -> 8B-aligned float2
    v2f av = *reinterpret_cast<const v2f*>(xrow + ka);   // X[m, ka], X[m, ka+1]
    v2f bv;
    bv.x = W[(size_t)ka * Hdim + n];            // W[ka,   n]
    bv.y = W[(size_t)(ka + 1) * Hdim + n];      // W[ka+1, n]
    c = __builtin_amdgcn_wmma_f32_16x16x4_f32(
        /*neg_a=*/false, av, /*neg_b=*/false, bv,
        /*c_mod=*/(short)0, c, /*reuse_a=*/false, /*reuse_b=*/false);
  }

  const float bias = (slab == 0) ? b1[n] : 0.0f;
  float* __restrict__ Out = (slab == 0) ? Aout : Bout;
  const int rowBase = tileM + ((lane >> 4) << 3);   // lanes 16-31 hold M+8
#pragma unroll
  for (int r = 0; r < 8; ++r) {
    Out[(size_t)(rowBase + r) * Hdim + n] = c[r] + bias;
  }
}

// ---------------------------------------------------------------------------
// Kernel 2: span logits + BCE-with-logits, tiled 16x16 (i,j) per block.
// logit[b,i,j] = sum_h gelu(A'[b,i,h] + Bm[b,j,h]) * W2[h] + b2
// grid = (L/16, L/16, B), block = 256.
// Inner loop uses float4 LDS reads (ds_load_b128): A-read covers all 64 banks
// conflict-free across 16 lanes (row stride 132 -> bank ti*4+h), B/W2 reads
// are broadcasts.
// ---------------------------------------------------------------------------
__global__ __launch_bounds__(256) void span_bce_kernel(
    const float* __restrict__ A, const float* __restrict__ Bm,
    const float* __restrict__ W2, const float* __restrict__ b2,
    const int* __restrict__ span_pos, float* __restrict__ partial)
{
  __shared__ float As[16][132];   // stride 132 floats: %64 == 4, 16B aligned rows
  __shared__ float Bs[16][132];
  __shared__ float w2s[128];
  __shared__ float red[256];

  const int b  = blockIdx.z;
  const int i0 = blockIdx.y * 16;
  const int j0 = blockIdx.x * 16;
  const int t  = threadIdx.x;
  const int ti = t & 15;          // i within tile
  const int tj = t >> 4;          // j within tile

  const float* __restrict__ Abase = A  + (size_t)(b * Ldim + i0) * Hdim;
  const float* __restrict__ Bbase = Bm + (size_t)(b * Ldim + j0) * Hdim;

  const int lr = t >> 4;          // loader row 0..15
  const int lc = (t & 15) * 8;    // loader col 0..120

  float acc = 0.0f;
  for (int h0 = 0; h0 < Hdim; h0 += 128) {
    __syncthreads();
    {
      const float4* sa = reinterpret_cast<const float4*>(Abase + (size_t)lr * Hdim + h0 + lc);
      const float4* sb = reinterpret_cast<const float4*>(Bbase + (size_t)lr * Hdim + h0 + lc);
      float4 a0 = sa[0], a1 = sa[1];
      float4 b0 = sb[0], b1v = sb[1];
      *reinterpret_cast<float4*>(&As[lr][lc])     = a0;
      *reinterpret_cast<float4*>(&As[lr][lc + 4]) = a1;
      *reinterpret_cast<float4*>(&Bs[lr][lc])     = b0;
      *reinterpret_cast<float4*>(&Bs[lr][lc + 4]) = b1v;
    }
    if (t < 128) w2s[t] = W2[h0 + t];
    __syncthreads();

#pragma unroll 2
    for (int h = 0; h < 128; h += 4) {
      float4 a4 = *reinterpret_cast<const float4*>(&As[ti][h]);
      float4 b4 = *reinterpret_cast<const float4*>(&Bs[tj][h]);
      float4 w4 = *reinterpret_cast<const float4*>(&w2s[h]);
      float x0 = a4.x + b4.x, x1 = a4.y + b4.y;
      float x2 = a4.z + b4.z, x3 = a4.w + b4.w;
      const float k = 0.70710678118654752f;
      float g0 = 0.5f * x0 * (1.0f + erff(x0 * k));
      float g1 = 0.5f * x1 * (1.0f + erff(x1 * k));
      float g2 = 0.5f * x2 * (1.0f + erff(x2 * k));
      float g3 = 0.5f * x3 * (1.0f + erff(x3 * k));
      acc = fmaf(g0, w4.x, acc);
      acc = fmaf(g1, w4.y, acc);
      acc = fmaf(g2, w4.z, acc);
      acc = fmaf(g3, w4.w, acc);
    }
  }

  const float logit = acc + b2[0];
  const float z = (float)span_pos[(size_t)(b * Ldim + i0 + ti) * Ldim + (j0 + tj)];
  const float bce = fmaxf(logit, 0.0f) - logit * z + log1pf(expf(-fabsf(logit)));

  red[t] = bce;
  __syncthreads();
#pragma unroll
  for (int s = 128; s > 0; s >>= 1) {
    if (t < s) red[t] += red[t + s];
    __syncthreads();
  }
  if (t == 0) {
    partial[(blockIdx.z * gridDim.y + blockIdx.y) * gridDim.x + blockIdx.x] = red[0];
  }
}

// ---------------------------------------------------------------------------
// Kernel 3: start/end 2-class CE per token. 1024 tokens, grid=4, block=256.
// Writes per-block sum of (start_ce + end_ce) contributions.
// ---------------------------------------------------------------------------
__global__ __launch_bounds__(256) void token_ce_kernel(
    const float* __restrict__ seq, const int* __restrict__ spos,
    const int* __restrict__ epos,
    const float* __restrict__ Ws, const float* __restrict__ bs,
    const float* __restrict__ We, const float* __restrict__ be,
    float* __restrict__ partial)
{
  __shared__ float red[256];
  const int t = blockIdx.x * 256 + threadIdx.x;   // token id = b*L + l
  const float* __restrict__ row = seq + (size_t)t * Hdim;

  float s0 = bs[0], s1 = bs[1], e0 = be[0], e1 = be[1];
#pragma unroll 4
  for (int k = 0; k < Hdim; ++k) {
    float x = row[k];
    s0 = fmaf(x, Ws[2 * k],     s0);
    s1 = fmaf(x, Ws[2 * k + 1], s1);
    e0 = fmaf(x, We[2 * k],     e0);
    e1 = fmaf(x, We[2 * k + 1], e1);
  }
  float ms  = fmaxf(s0, s1);
  float lses = ms + logf(expf(s0 - ms) + expf(s1 - ms));
  float me  = fmaxf(e0, e1);
  float lsee = me + logf(expf(e0 - me) + expf(e1 - me));
  float sl = ((spos[t] == 1) ? s1 : s0) - lses;
  float el = ((epos[t] == 1) ? e1 : e0) - lsee;
  float contrib = -(sl + el);

  red[threadIdx.x] = contrib;
  __syncthreads();
#pragma unroll
  for (int s = 128; s > 0; s >>= 1) {
    if (threadIdx.x < s) red[threadIdx.x] += red[threadIdx.x + s];
    __syncthreads();
  }
  if (threadIdx.x == 0) partial[blockIdx.x] = red[0];
}

// ---------------------------------------------------------------------------
// Kernel 4: deterministic fixed-order finalize.
// ---------------------------------------------------------------------------
__global__ void finalize_kernel(const float* __restrict__ spanP, int nSpan,
                                const float* __restrict__ tokP, int nTok,
                                float* __restrict__ out)
{
  float ssum = 0.0f;
  for (int i = 0; i < nSpan; ++i) ssum += spanP[i];
  float tsum = 0.0f;
  for (int i = 0; i < nTok; ++i) tsum += tokP[i];
  const float span_loss = ssum / (float)(Bdim * Ldim * Ldim);
  const float tok_loss  = tsum / (float)(Bdim * Ldim);   // = start_loss + end_loss
  out[0] = tok_loss + span_loss;                         // W_START=W_END=W_SPAN=1
}

extern "C" void kernel_launch(void* const* d_in, const int* in_sizes, int n_in,
                              void* d_out, int out_size, void* d_ws, size_t ws_size,
                              hipStream_t stream) {
  const float* seq    = (const float*)d_in[0];
  const int*   spos   = (const int*)  d_in[1];
  const int*   epos   = (const int*)  d_in[2];
  const int*   spanp  = (const int*)  d_in[3];
  const float* Wstart = (const float*)d_in[4];
  const float* bstart = (const float*)d_in[5];
  const float* Wend   = (const float*)d_in[6];
  const float* bend   = (const float*)d_in[7];
  const float* W1     = (const float*)d_in[8];
  const float* b1     = (const float*)d_in[9];
  const float* W2     = (const float*)d_in[10];
  const float* b2     = (const float*)d_in[11];

  float* ws    = (float*)d_ws;
  float* Aout  = ws;                       // 1024*768 floats
  float* Bout  = ws + 786432;              // 1024*768 floats
  float* spanP = ws + 1572864;             // 512 floats
  float* tokP  = spanP + 512;              // 4 floats

  dim3 gGemm(Hdim / 64, (Bdim * Ldim) / 32, 2);   // (12, 32, 2)
  gemm_proj_wmma<<<gGemm, 256, 0, stream>>>(seq, W1, b1, Aout, Bout);

  token_ce_kernel<<<4, 256, 0, stream>>>(seq, spos, epos, Wstart, bstart,
                                         Wend, bend, tokP);

  dim3 gSpan(Ldim / 16, Ldim / 16, Bdim);         // (8, 8, 8) = 512 blocks
  span_bce_kernel<<<gSpan, 256, 0, stream>>>(Aout, Bout, W2, b2, spanp, spanP);

  finalize_kernel<<<1, 1, 0, stream>>>(spanP, 512, tokP, 4, (float*)d_out);
}